// PointTransformerLayerMLPPooling_75093208203620
// MI455X (gfx1250) — compile-verified
//
#include <hip/hip_runtime.h>
#include <hip/hip_bf16.h>
#include <float.h>
#include <stdint.h>

typedef __attribute__((ext_vector_type(2))) float v2f;
typedef __attribute__((ext_vector_type(8))) float v8f;

#define BN_EPS 1e-5f
#define K_NN 16

// GEMM tiling
#define BM 128           // rows per workgroup (8 waves x 16)
#define BNT 64           // cols per workgroup (4 WMMA groups of 16)
#define BK 32            // K slab
#define A_STRIDE 36      // 32 + 4 pad: 16B-aligned rows, conflict-free b64 frag reads
#define WP_STRIDE 160    // (64 cols x 2) + 32 pad: half-waves hit disjoint bank halves

// ---------------------------------------------------------------------------
// Kernel 1: fused Linear + BatchNorm(inference) + ReLU.
// fp32 WMMA 16x16x4. A: double-buffered async global->LDS (ASYNCcnt).
// W: global->VGPR->LDS, k-pair interleaved so each B fragment is one b64.
// ---------------------------------------------------------------------------
__global__ __launch_bounds__(256) void mlp_gemm_bn_relu(
    const float* __restrict__ x, const float* __restrict__ W,
    const float* __restrict__ bias, const float* __restrict__ gamma,
    const float* __restrict__ beta, const float* __restrict__ rmean,
    const float* __restrict__ rvar, float* __restrict__ h,
    int N, int Cin, int Cout)
{
    __shared__ float sA[2][BM * A_STRIDE];      // 2 x 18KB
    __shared__ float sWp[2][(BK / 2) * WP_STRIDE]; // 2 x 10KB, k-pair interleaved

    const int tid  = threadIdx.x;
    const int lane = tid & 31;
    const int wave = tid >> 5;
    const int half = lane >> 4;      // 0: lanes 0-15, 1: lanes 16-31
    const int l16  = lane & 15;

    const int mBase = blockIdx.x * BM;
    const int n0    = blockIdx.y * BNT;
    const int mw    = wave * 16;

    v8f acc[4] = {v8f{}, v8f{}, v8f{}, v8f{}};

    // A slab: 128 rows x 32 K, async global->LDS, 4 b128 per thread.
    auto loadA = [&](int buf, int kb) {
#pragma unroll
        for (int i = 0; i < 4; ++i) {
            const int id = tid + 256 * i;          // 0..1023
            const int r  = id >> 3;                // row 0..127
            const int c  = id & 7;                 // float4 chunk 0..7
            const uint32_t lds = (uint32_t)(uintptr_t)&sA[buf][r * A_STRIDE + c * 4];
            const uint64_t ga  = (uint64_t)(uintptr_t)(x + (size_t)(mBase + r) * Cin + kb + c * 4);
            asm volatile("global_load_async_to_lds_b128 %0, %1, off"
                         :: "v"(lds), "v"(ga) : "memory");
        }
    };

    // W slab: 16 k-pairs x 64 cols; each thread owns 4 (kp,col) pairs.
    auto loadWregs = [&](int kb, float (&wreg)[4][2]) {
#pragma unroll
        for (int i = 0; i < 4; ++i) {
            const int id = tid + 256 * i;          // 0..1023
            const int kp = id >> 6;                // 0..15
            const int c  = id & 63;                // 0..63
            const float* g0 = W + (size_t)(kb + 2 * kp) * Cout + n0 + c;
            wreg[i][0] = g0[0];
            wreg[i][1] = g0[Cout];
        }
    };
    auto storeWregs = [&](int buf, const float (&wreg)[4][2]) {
#pragma unroll
        for (int i = 0; i < 4; ++i) {
            const int id = tid + 256 * i;
            const int kp = id >> 6;
            const int c  = id & 63;
            v2f v; v[0] = wreg[i][0]; v[1] = wreg[i][1];
            *(v2f*)&sWp[buf][kp * WP_STRIDE + c * 2] = v;   // ds_store_b64
        }
    };

    {   // prologue: fill buffer 0
        loadA(0, 0);
        float wreg[4][2];
        loadWregs(0, wreg);
        storeWregs(0, wreg);
        asm volatile("s_wait_asynccnt 0x0" ::: "memory");
        __syncthreads();
    }

    const int nSlab = Cin / BK;
    for (int s = 0; s < nSlab; ++s) {
        const int cur = s & 1;
        const bool pre = (s + 1 < nSlab);
        float wreg[4][2];
        if (pre) {
            loadA(cur ^ 1, (s + 1) * BK);          // async, overlaps compute
            loadWregs((s + 1) * BK, wreg);         // global->regs, overlaps compute
        }

        // A frag: row mw+l16, K offset 2*half; W frag: kp = kk/2 + half, one b64.
        const float* __restrict__ aBase = &sA[cur][(mw + l16) * A_STRIDE + (half << 1)];
        const float* __restrict__ wBase = &sWp[cur][half * WP_STRIDE + l16 * 2];

#pragma unroll
        for (int kk = 0; kk < BK; kk += 4) {
            const v2f a = *(const v2f*)(aBase + kk);
            const float* wk = wBase + (kk >> 1) * WP_STRIDE;
#pragma unroll
            for (int g = 0; g < 4; ++g) {
                const v2f bf = *(const v2f*)(wk + g * 32);
                acc[g] = __builtin_amdgcn_wmma_f32_16x16x4_f32(
                    false, a, false, bf, (short)0, acc[g], false, false);
            }
        }

        if (pre) storeWregs(cur ^ 1, wreg);        // after compute on 'cur'
        asm volatile("s_wait_asynccnt 0x0" ::: "memory");
        __syncthreads();
    }

    // Epilogue: (acc + b - mean) * gamma*rsqrt(var+eps) + beta, ReLU -> h.
#pragma unroll
    for (int g = 0; g < 4; ++g) {
        const int col    = n0 + g * 16 + l16;
        const float sc   = gamma[col] * rsqrtf(rvar[col] + BN_EPS);
        const float base = (bias[col] - rmean[col]) * sc + beta[col];
        float* __restrict__ out = h + (size_t)(mBase + mw + half * 8) * Cout + col;
#pragma unroll
        for (int i = 0; i < 8; ++i) {
            out[(size_t)i * Cout] = fmaxf(fmaf(acc[g][i], sc, base), 0.0f);
        }
    }
}

// ---------------------------------------------------------------------------
// Kernel 2: per-scene kNN (k=16, includes self). One query per thread; LDS
// candidate tiles; register-resident top-16 with guarded unrolled insertion.
// ---------------------------------------------------------------------------
__global__ __launch_bounds__(256) void knn_kernel(
    const float* __restrict__ p, const int* __restrict__ o,
    int* __restrict__ nn, int N, int B)
{
    __shared__ float sx[256], sy[256], sz[256];
    const int tid = threadIdx.x;
    const int q   = blockIdx.x * 256 + tid;
    const bool valid = q < N;

    float qx = 0.f, qy = 0.f, qz = 0.f;
    if (valid) { qx = p[3 * q + 0]; qy = p[3 * q + 1]; qz = p[3 * q + 2]; }

    int myStart = 0;
    {
        int s0 = 0;
        for (int s = 0; s < B; ++s) { int e = o[s]; if (q < e) { myStart = s0; break; } s0 = e; }
    }

    float bestd[K_NN];
    int   besti[K_NN];
#pragma unroll
    for (int j = 0; j < K_NN; ++j) { bestd[j] = FLT_MAX; besti[j] = q; }

    const int qb0 = blockIdx.x * 256;
    const int qb1 = min(qb0 + 256, N);

    int start = 0;
    for (int s = 0; s < B; ++s) {
        const int end = o[s];
        if (end > qb0 && start < qb1) {          // block-uniform branch
            for (int t = start; t < end; t += 256) {
                __syncthreads();
                const int c = t + tid;
                if (c < end) {
                    sx[tid] = p[3 * c + 0];
                    sy[tid] = p[3 * c + 1];
                    sz[tid] = p[3 * c + 2];
                }
                __syncthreads();
                const int nc = min(256, end - t);
                if (valid && myStart == start) {
                    for (int j = 0; j < nc; ++j) {
                        const float dx = qx - sx[j];
                        const float dy = qy - sy[j];
                        const float dz = qz - sz[j];
                        const float d  = fmaf(dx, dx, fmaf(dy, dy, dz * dz));
                        if (d < bestd[K_NN - 1]) {
                            float dd = d; int ii = t + j;
#pragma unroll
                            for (int k = 0; k < K_NN; ++k) {
                                if (dd < bestd[k]) {
                                    float tf = bestd[k]; bestd[k] = dd; dd = tf;
                                    int   ti = besti[k]; besti[k] = ii; ii = ti;
                                }
                            }
                        }
                    }
                }
            }
        }
        start = end;
    }

    if (valid) {
#pragma unroll
        for (int j = 0; j < K_NN; ++j) nn[q * K_NN + j] = besti[j];
    }
}

// ---------------------------------------------------------------------------
// Kernel 3: gather 16 neighbor rows + channel-wise max. L2-resident h.
// ---------------------------------------------------------------------------
__global__ void group_max_pool(const float* __restrict__ h,
                               const int* __restrict__ nn,
                               float* __restrict__ out, int Cout)
{
    const int q = blockIdx.x;
    const int t = threadIdx.x;               // Cout/4 threads
    const int* __restrict__ idx = nn + q * K_NN;

    float4 m = make_float4(-FLT_MAX, -FLT_MAX, -FLT_MAX, -FLT_MAX);
#pragma unroll
    for (int j = 0; j < K_NN; ++j) {
        const float4 v = ((const float4*)(h + (size_t)idx[j] * Cout))[t];
        m.x = fmaxf(m.x, v.x);
        m.y = fmaxf(m.y, v.y);
        m.z = fmaxf(m.z, v.z);
        m.w = fmaxf(m.w, v.w);
    }
    ((float4*)(out + (size_t)q * Cout))[t] = m;
}

// ---------------------------------------------------------------------------
extern "C" void kernel_launch(void* const* d_in, const int* in_sizes, int n_in,
                              void* d_out, int out_size, void* d_ws, size_t ws_size,
                              hipStream_t stream)
{
    const float* p     = (const float*)d_in[0];
    const float* x     = (const float*)d_in[1];
    const int*   o     = (const int*)  d_in[2];
    const float* W     = (const float*)d_in[3];
    const float* b     = (const float*)d_in[4];
    const float* gamma = (const float*)d_in[5];
    const float* beta  = (const float*)d_in[6];
    const float* rmean = (const float*)d_in[7];
    const float* rvar  = (const float*)d_in[8];

    const int N    = in_sizes[0] / 3;        // 16384
    const int Cin  = in_sizes[1] / N;        // 256
    const int B    = in_sizes[2];            // 4
    const int Cout = in_sizes[3] / Cin;      // 512

    float* h  = (float*)d_ws;                                           // N*Cout f32
    int*   nn = (int*)((char*)d_ws + (size_t)N * Cout * sizeof(float)); // N*16 i32

    dim3 g1(N / BM, Cout / BNT);
    mlp_gemm_bn_relu<<<g1, 256, 0, stream>>>(x, W, b, gamma, beta, rmean, rvar,
                                             h, N, Cin, Cout);

    knn_kernel<<<(N + 255) / 256, 256, 0, stream>>>(p, o, nn, N, B);

    group_max_pool<<<N, Cout / 4, 0, stream>>>(h, nn, (float*)d_out, Cout);
}